// GraphConvolutionLayer_37864431681684
// MI455X (gfx1250) — compile-verified
//
#include <hip/hip_runtime.h>

// GraphConv: out = segment_sum((x@W)[src] * (edge_attr@edge_w), dst) + b
//
// MI455X plan:
//  - Both GEMMs on the matrix pipe via V_WMMA_F32_16X16X4_F32 (exact f32).
//  - B operands staged in LDS in K-pair-interleaved layout so each WMMA B
//    fragment is ONE ds_load_b64 into an even-aligned VGPR pair (no v_mov
//    marshalling, half the DS ops vs. per-row layout).
//  - ew [800k x 128] never materialized: edge GEMM fused with
//    gather/modulate/scatter. xw (51MB) + out (51MB) stay L2-resident (192MB
//    L2); streamed edge data uses non-temporal loads so it doesn't evict them.
//  - Scatter-add via global_atomic_add_f32 (no-return -> STOREcnt path).
//  - d_ws holds xw: n_nodes*128*4 = 51.2 MB.

typedef __attribute__((ext_vector_type(2))) float v2f;
typedef __attribute__((ext_vector_type(8))) float v8f;

#define DD   128   // D_IN == D_OUT == 128
#define EDIM 16

// ---------------------------------------------------------------- init: out = b
__global__ __launch_bounds__(256) void init_bias_kernel(float* __restrict__ out,
                                                        const float* __restrict__ b,
                                                        long total) {
  long i = (long)blockIdx.x * blockDim.x + threadIdx.x;
  if (i < total) out[i] = b[i & (DD - 1)];
}

// ------------------------------------------------------- xw = x @ W  (WMMA f32)
// Block = 128 threads (4 waves). Each wave: 16 rows x 128 cols.
// A frag (16x4 f32): lanes 0-15 hold M=lane, K=k+0/k+1; lanes 16-31 M=lane-16,
// K=k+2/k+3 -> contiguous float2 loads.
// B frag: lanes 0-15 hold N=lane, K=k+0/k+1; lanes 16-31 K=k+2/k+3 -> one
// ds_load_b64 from the pair-interleaved LDS image of W.
__global__ __launch_bounds__(128) void node_gemm_wmma(const float* __restrict__ x,
                                                      const float* __restrict__ W,
                                                      float* __restrict__ xw,
                                                      int n_nodes) {
  // Ws[(kp*DD + n)*2 + j] = W[2*kp + j][n],  kp = 0..63, j = 0..1
  __shared__ float Ws[(DD / 2) * DD * 2];  // 64 KB (WGP has 320 KB)
  int tid = threadIdx.x;
  for (int i = tid * 4; i < (DD / 2) * DD; i += 128 * 4) {
    int kp = i / DD, n = i % DD;
    float4 r0 = *(const float4*)&W[(2 * kp + 0) * DD + n];
    float4 r1 = *(const float4*)&W[(2 * kp + 1) * DD + n];
    float4 p0 = {r0.x, r1.x, r0.y, r1.y};
    float4 p1 = {r0.z, r1.z, r0.w, r1.w};
    *(float4*)&Ws[(kp * DD + n) * 2 + 0] = p0;
    *(float4*)&Ws[(kp * DD + n) * 2 + 4] = p1;
  }
  __syncthreads();

  const int wave = tid >> 5, lane = tid & 31;
  const int half = lane >> 4, lm = lane & 15;
  const long row0 = (long)blockIdx.x * 64 + wave * 16;

  long ar = row0 + lm;                       // clamp so EXEC stays all-ones
  if (ar > (long)n_nodes - 1) ar = (long)n_nodes - 1;
  const float* xrow = x + ar * DD;

  v8f acc[8];
#pragma unroll
  for (int t = 0; t < 8; ++t)
#pragma unroll
    for (int v = 0; v < 8; ++v) acc[t][v] = 0.0f;

  for (int k = 0; k < DD; k += 4) {
    v2f a = *(const v2f*)&xrow[k + half * 2];
    int kp = (k >> 1) + half;                // K-pair index for this half-wave
#pragma unroll
    for (int t = 0; t < 8; ++t) {
      int n = t * 16 + lm;
      v2f bf = *(const v2f*)&Ws[(kp * DD + n) * 2];   // single ds_load_b64
      acc[t] = __builtin_amdgcn_wmma_f32_16x16x4_f32(false, a, false, bf,
                                                     (short)0, acc[t], false, false);
    }
  }

  // C layout: VGPR v -> row (v + half*8), col = t*16 + lm
#pragma unroll
  for (int t = 0; t < 8; ++t) {
    int n = t * 16 + lm;
#pragma unroll
    for (int v = 0; v < 8; ++v) {
      long m = row0 + v + half * 8;
      if (m < n_nodes) xw[m * DD + n] = acc[t][v];
    }
  }
}

// ------------------- fused: ew-GEMM (WMMA) + gather xw[src] + scatter-add out[dst]
// Block = 256 threads (8 waves); each wave owns a group of 16 edges.
__global__ __launch_bounds__(256) void edge_fused_wmma(const float* __restrict__ edge_attr,
                                                       const int* __restrict__ edge_index,
                                                       const float* __restrict__ xw,
                                                       const float* __restrict__ edge_w,
                                                       float* __restrict__ out,
                                                       long n_edges) {
  // EWs[(kp*DD + n)*2 + j] = edge_w[2*kp + j][n],  kp = 0..7
  __shared__ float EWs[(EDIM / 2) * DD * 2];   // 8 KB
  __shared__ int   sIdx[8][2][16];             // per-wave src/dst staging

  int tid = threadIdx.x;
  const int wave = tid >> 5, lane = tid & 31;
  const int half = lane >> 4, lm = lane & 15;
  const long e0 = ((long)blockIdx.x * 8 + wave) * 16;

  for (int i = tid * 4; i < (EDIM / 2) * DD; i += 256 * 4) {
    int kp = i / DD, n = i % DD;
    float4 r0 = *(const float4*)&edge_w[(2 * kp + 0) * DD + n];
    float4 r1 = *(const float4*)&edge_w[(2 * kp + 1) * DD + n];
    float4 p0 = {r0.x, r1.x, r0.y, r1.y};
    float4 p1 = {r0.z, r1.z, r0.w, r1.w};
    *(float4*)&EWs[(kp * DD + n) * 2 + 0] = p0;
    *(float4*)&EWs[(kp * DD + n) * 2 + 4] = p1;
  }
  if (lane < 16) {
    long e = e0 + lane;
    if (e > n_edges - 1) e = n_edges - 1;
    sIdx[wave][0][lane] = __builtin_nontemporal_load(&edge_index[e]);            // src
    sIdx[wave][1][lane] = __builtin_nontemporal_load(&edge_index[n_edges + e]);  // dst
  }
  __syncthreads();

  long ae = e0 + lm;
  if (ae > n_edges - 1) ae = n_edges - 1;
  const float* arow = edge_attr + ae * EDIM;   // 16 floats, 64B aligned

  v8f acc[8];
#pragma unroll
  for (int t = 0; t < 8; ++t)
#pragma unroll
    for (int v = 0; v < 8; ++v) acc[t][v] = 0.0f;

#pragma unroll
  for (int kk = 0; kk < 4; ++kk) {             // K = 16 -> 4 WMMA k-steps
    int k = kk * 4;
    v2f a = __builtin_nontemporal_load((const v2f*)&arow[k + half * 2]);
    int kp = (k >> 1) + half;
#pragma unroll
    for (int t = 0; t < 8; ++t) {
      int n = t * 16 + lm;
      v2f bf = *(const v2f*)&EWs[(kp * DD + n) * 2];  // single ds_load_b64
      acc[t] = __builtin_amdgcn_wmma_f32_16x16x4_f32(false, a, false, bf,
                                                     (short)0, acc[t], false, false);
    }
  }

  // acc[t][v] = ew value for edge (e0 + v + half*8), column t*16+lm.
  // Gather xw[src], modulate, atomic scatter-add into out[dst].
#pragma unroll
  for (int v = 0; v < 8; ++v) {
    int m = v + half * 8;
    if (e0 + m < n_edges) {
      long sbase = (long)sIdx[wave][0][m] * DD;
      long dbase = (long)sIdx[wave][1][m] * DD;
#pragma unroll
      for (int t = 0; t < 8; ++t) {
        int n = t * 16 + lm;
        float msg = acc[t][v] * xw[sbase + n];   // L2-resident gather, 64B/half-wave
        atomicAdd(&out[dbase + n], msg);         // global_atomic_add_f32 (no return)
      }
    }
  }
}

// --------------------------------------------------------------------- launcher
extern "C" void kernel_launch(void* const* d_in, const int* in_sizes, int n_in,
                              void* d_out, int out_size, void* d_ws, size_t ws_size,
                              hipStream_t stream) {
  const float* x  = (const float*)d_in[0];
  const int*   ei = (const int*)d_in[1];   // integer input -> const int* per contract
  const float* ea = (const float*)d_in[2];
  const float* W  = (const float*)d_in[3];
  const float* ew = (const float*)d_in[4];
  const float* b  = (const float*)d_in[5];

  const int  n_nodes = in_sizes[0] / DD;
  const long n_edges = (long)in_sizes[2] / EDIM;

  float* xw  = (float*)d_ws;               // n_nodes*128*4 = 51.2 MB scratch
  float* out = (float*)d_out;

  const long total = (long)n_nodes * DD;
  init_bias_kernel<<<(int)((total + 255) / 256), 256, 0, stream>>>(out, b, total);

  node_gemm_wmma<<<(n_nodes + 63) / 64, 128, 0, stream>>>(x, W, xw, n_nodes);

  const long groups = (n_edges + 15) / 16;
  edge_fused_wmma<<<(int)((groups + 7) / 8), 256, 0, stream>>>(ea, ei, xw, ew, out,
                                                               n_edges);
}